// SPABlock_4784593567750
// MI455X (gfx1250) — compile-verified
//
#include <hip/hip_runtime.h>
#include <hip/hip_bf16.h>

// Problem constants (x: [B, C, H, W] fp32, out: [B, K, C] fp32)
#define BATCH 16
#define CHANS 64
#define NPOS  65536   // 256*256
#define TOPK  64

typedef __attribute__((ext_vector_type(2))) float v2f;
typedef __attribute__((ext_vector_type(8))) float v8f;

// ---------------------------------------------------------------------------
// Kernel 1: per-position squared-L2 norms via V_WMMA_F32_16X16X4_F32.
// One wave handles 16 consecutive positions n0..n0+15 of one batch.
// Gram trick: D = A * A^T accumulated over 16 chunks of 4 channels;
// diag(D) = sum_c x[b,c,n]^2. For fp32 16x16x4, the A (16x4, MxK) and
// B (4x16, KxN) per-lane VGPR layouts coincide for the symmetric case:
//   lane L: rows/cols index = L&15, channels {2*(L>>4), 2*(L>>4)+1}
// so the same v2f register pair serves as both A and B.
// ---------------------------------------------------------------------------
__global__ void __launch_bounds__(256)
spab_norms_wmma(const float* __restrict__ x, float* __restrict__ norms) {
    const int lane  = threadIdx.x & 31;
    const int gwave = blockIdx.x * 8 + (threadIdx.x >> 5);  // 8 waves / block
    const int b  = gwave >> 12;            // 4096 groups of 16 positions per batch
    const int n0 = (gwave & 4095) << 4;

    const int m = lane & 15;               // position within group (M and N index)
    const int h = lane >> 4;               // lane-half selects channel pair {2h, 2h+1}

    const float* base = x + (size_t)b * CHANS * NPOS + (size_t)n0 + m;

#if __has_builtin(__builtin_amdgcn_wmma_f32_16x16x4_f32)
    v8f acc = {};
#pragma unroll
    for (int chunk = 0; chunk < 16; ++chunk) {
        const float* p = base + (size_t)(chunk * 4 + 2 * h) * NPOS;
        v2f a;
        a.x = p[0];             // channel chunk*4 + 2h
        a.y = p[NPOS];          // channel chunk*4 + 2h + 1
        // D += A * A^T  (same registers for A and B by layout symmetry)
        acc = __builtin_amdgcn_wmma_f32_16x16x4_f32(
            /*neg_a=*/false, a, /*neg_b=*/false, a,
            /*c_mod=*/(short)0, acc, /*reuse_a=*/false, /*reuse_b=*/false);
    }
    // Diagonal of 16x16 f32 D:  D[M=v+8*(L>>4), N=L&15] in VGPR v of lane L.
    // M==N holds for lanes 0..7 (v=lane) and lanes 24..31 (v=lane-24).
    float vals[8];
#pragma unroll
    for (int i = 0; i < 8; ++i) vals[i] = acc[i];
    const int sel = (lane < 8) ? lane : (lane - 24);
    float diag = 0.0f;
#pragma unroll
    for (int i = 0; i < 8; ++i)
        if (sel == i) diag = vals[i];

    float* nb = norms + (size_t)b * NPOS + n0;
    if (lane < 8)        nb[lane]      = diag;
    else if (lane >= 24) nb[lane - 16] = diag;
#else
    // Fallback (no f32 WMMA builtin): lanes 0..15 each reduce one position.
    if (lane < 16) {
        float s = 0.0f;
        const float* p = base - m + (n0 >= 0 ? 0 : 0) + (lane & 15);
        p = x + (size_t)b * CHANS * NPOS + (size_t)n0 + lane;
#pragma unroll 8
        for (int c = 0; c < CHANS; ++c) {
            float v = p[(size_t)c * NPOS];
            s = fmaf(v, v, s);
        }
        norms[(size_t)b * NPOS + n0 + lane] = s;
    }
#endif
}

// ---------------------------------------------------------------------------
// Kernel 2: exact top-64 (sorted desc, ties -> lower index) + fused gather.
// One 1024-thread block per batch. Each thread caches 64 norm bit-patterns
// in registers (norms >= 0 so IEEE bits are order-monotone). 64 rounds of
// block-wide argmax on keys (valbits<<32 | ~idx); winner's owner zaps it.
// ---------------------------------------------------------------------------
__global__ void __launch_bounds__(1024)
spab_topk_gather(const float* __restrict__ x, const float* __restrict__ norms,
                 float* __restrict__ out) {
    const int b = blockIdx.x;
    const int t = threadIdx.x;

    __shared__ unsigned long long warp_best[32];
    __shared__ unsigned long long bcast;
    __shared__ unsigned int sel_idx[TOPK];

    const float* nb = norms + (size_t)b * NPOS;
    unsigned int vbits[64];
#pragma unroll
    for (int j = 0; j < 64; ++j)
        vbits[j] = __float_as_uint(nb[j * 1024 + t]);

    for (int k = 0; k < TOPK; ++k) {
        // local argmax over this thread's 64 slots
        unsigned long long best = 0ull;
#pragma unroll
        for (int j = 0; j < 64; ++j) {
            const unsigned int n = (unsigned int)(j * 1024 + t);
            const unsigned long long key =
                ((unsigned long long)vbits[j] << 32) | (unsigned int)(~n);
            best = (key > best) ? key : best;
        }
        // wave32 reduction
#pragma unroll
        for (int off = 16; off > 0; off >>= 1) {
            unsigned long long o = __shfl_down(best, off, 32);
            best = (o > best) ? o : best;
        }
        if ((t & 31) == 0) warp_best[t >> 5] = best;
        __syncthreads();
        if (t < 32) {
            unsigned long long v = warp_best[t];
#pragma unroll
            for (int off = 16; off > 0; off >>= 1) {
                unsigned long long o = __shfl_down(v, off, 32);
                v = (o > v) ? o : v;
            }
            if (t == 0) {
                bcast = v;
                sel_idx[k] = ~(unsigned int)v;   // recover index
            }
        }
        __syncthreads();
        const unsigned int win = ~(unsigned int)bcast;
        // owner removes the winner from its register cache
        if ((win & 1023u) == (unsigned int)t) {
            const int jz = (int)(win >> 10);
#pragma unroll
            for (int j = 0; j < 64; ++j)
                if (j == jz) vbits[j] = 0u;
        }
        __syncthreads();
    }

    // fused gather: out[b, k, c] = x[b, c, sel_idx[k]]
    for (int i = t; i < TOPK * CHANS; i += 1024) {
        const int k = i >> 6;
        const int c = i & (CHANS - 1);
        out[(size_t)b * (TOPK * CHANS) + i] =
            x[((size_t)(b * CHANS + c)) * NPOS + sel_idx[k]];
    }
}

// ---------------------------------------------------------------------------
extern "C" void kernel_launch(void* const* d_in, const int* in_sizes, int n_in,
                              void* d_out, int out_size, void* d_ws, size_t ws_size,
                              hipStream_t stream) {
    (void)in_sizes; (void)n_in; (void)out_size; (void)ws_size;
    const float* x = (const float*)d_in[0];
    float* out   = (float*)d_out;
    float* norms = (float*)d_ws;            // needs B*NPOS*4 = 4 MB of scratch

    // Pass 1: streaming norm pass (bandwidth-bound; 268 MB ~= 11.5 us @ 23.3 TB/s)
    // waves = B * NPOS/16 = 65536 -> 8192 blocks of 256 threads (8 waves each)
    spab_norms_wmma<<<dim3(8192), dim3(256), 0, stream>>>(x, norms);

    // Pass 2: per-batch top-64 from L2-resident norms + gather
    spab_topk_gather<<<dim3(BATCH), dim3(1024), 0, stream>>>(x, norms, out);
}